// TtMistralAttention_979252544185
// MI455X (gfx1250) — compile-verified
//
#include <hip/hip_runtime.h>
#include <hip/hip_bf16.h>
#include <math.h>

// ---------------------------------------------------------------------------
// Shapes (from the reference)
// ---------------------------------------------------------------------------
#define HIDDEN     4096
#define N_HEADS    32
#define N_KV       8
#define HEAD_DIM   128
#define BATCH      32
#define CACHE_LEN  512
#define QKV_F      (HIDDEN + 2 * N_KV * HEAD_DIM)   // 6144
#define V_OFF      (HIDDEN + N_KV * HEAD_DIM)       // 5120
#define G          (N_HEADS / N_KV)                 // 4

typedef __attribute__((ext_vector_type(2))) float v2f;
typedef __attribute__((ext_vector_type(8))) float v8f;

// ---------------------------------------------------------------------------
// fp32 WMMA GEMM, M = 32 fixed:  D(32 x N) = A(32 x K) * B(K x N)
//   One wave owns a 16-wide column strip and BOTH 16-row M-tiles, so each
//   B fragment is loaded once and fed to two V_WMMA_F32_16X16X4_F32 ops.
//   Weights (B) are streamed exactly once -> bandwidth-optimal.
//   block = (32, WAVES_N) ; grid = (N / 16 / WAVES_N)
// ---------------------------------------------------------------------------
#define WAVES_N 8

__global__ __launch_bounds__(32 * WAVES_N)
void wmma_gemm_f32_m32(const float* __restrict__ A, const float* __restrict__ B,
                       float* __restrict__ D, int N, int K,
                       int lda, int ldb, int ldd) {
    const int lane   = threadIdx.x;          // 0..31 (wave32)
    const int half   = lane >> 4;            // 0: lanes 0-15, 1: lanes 16-31
    const int ml     = lane & 15;
    const int n_tile = blockIdx.x * WAVES_N + threadIdx.y;   // wave-uniform
    if (n_tile * 16 >= N) return;            // uniform per wave -> EXEC stays full

    const int n = n_tile * 16 + ml;          // B/D column this lane feeds

    // Pointer-incremented streams (no per-iteration index multiplies).
    const float* Ap0 = A + (size_t)ml        * lda + 2 * half;   // rows 0..15
    const float* Ap1 = A + (size_t)(ml + 16) * lda + 2 * half;   // rows 16..31
    const float* Bp  = B + (size_t)(2 * half) * ldb + n;
    const size_t bstep = 4 * (size_t)ldb;

    v8f c0 = {};   // D rows 0..15
    v8f c1 = {};   // D rows 16..31

    for (int k = 0; k < K; k += 4) {
        v2f a0 = *(const v2f*)Ap0;           // A[m   ][k+2h .. k+2h+1]
        v2f a1 = *(const v2f*)Ap1;           // A[m+16][k+2h .. k+2h+1]
        v2f b;
        b.x = Bp[0];                         // B[k+2h  ][n]
        b.y = Bp[ldb];                       // B[k+2h+1][n]

        if ((k & 31) == 0 && (k + 32) < K)   // stream-ahead hint for weights
            __builtin_prefetch(Bp + 8 * bstep, 0, 1);

        c0 = __builtin_amdgcn_wmma_f32_16x16x4_f32(
                 false, a0, false, b, (short)0, c0, false, false);
        c1 = __builtin_amdgcn_wmma_f32_16x16x4_f32(
                 false, a1, false, b, (short)0, c1, false, false);

        Ap0 += 4; Ap1 += 4; Bp += bstep;
    }

    // D fragment: VGPR i -> row (i + 8*half), col = n
    float* D0 = D + (size_t)(8 * half)      * ldd + n;
    float* D1 = D + (size_t)(16 + 8 * half) * ldd + n;
    #pragma unroll
    for (int i = 0; i < 8; ++i) {
        D0[(size_t)i * ldd] = c0[i];
        D1[(size_t)i * ldd] = c1[i];
    }
}

// ---------------------------------------------------------------------------
// RoPE as a gathered WMMA GEMM:
//   rows 0..1023   : q  row r = h*32+b   -> xqkv[b][h*128 + c]
//   rows 1024..1279: k  row r' = kv*32+b -> xqkv[b][4096 + kv*128 + c]
//   R = rows * rot_mat (128x128).  80 m-tiles x 8 n-tiles.
// ---------------------------------------------------------------------------
__device__ __forceinline__ size_t qkv_row_base(int r) {
    if (r < N_HEADS * BATCH) {
        int h = r >> 5, b = r & 31;
        return (size_t)b * QKV_F + (size_t)h * HEAD_DIM;
    }
    int rk = r - N_HEADS * BATCH;
    int kv = rk >> 5, b = rk & 31;
    return (size_t)b * QKV_F + HIDDEN + (size_t)kv * HEAD_DIM;
}

__global__ __launch_bounds__(32 * WAVES_N)
void rope_wmma(const float* __restrict__ xqkv, const float* __restrict__ rot,
               float* __restrict__ q_rope, float* __restrict__ k_new) {
    const int lane   = threadIdx.x;
    const int half   = lane >> 4;
    const int ml     = lane & 15;
    const int n_tile = threadIdx.y;          // 8 waves cover 128 columns
    const int m_tile = blockIdx.x;           // 0..79

    const int r = m_tile * 16 + ml;
    const int n = n_tile * 16 + ml;

    const float* Ap = xqkv + qkv_row_base(r) + 2 * half;
    const float* Bp = rot + (size_t)(2 * half) * HEAD_DIM + n;

    v8f c = {};
    #pragma unroll 4
    for (int k = 0; k < HEAD_DIM; k += 4) {
        v2f a = *(const v2f*)Ap;
        v2f b;
        b.x = Bp[0];
        b.y = Bp[HEAD_DIM];
        c = __builtin_amdgcn_wmma_f32_16x16x4_f32(
                false, a, false, b, (short)0, c, false, false);
        Ap += 4; Bp += 4 * HEAD_DIM;
    }

    #pragma unroll
    for (int i = 0; i < 8; ++i) {
        int row = m_tile * 16 + i + 8 * half;     // tiles never straddle q/k split
        if (row < N_HEADS * BATCH)
            q_rope[(size_t)row * HEAD_DIM + n] = c[i];
        else
            k_new[(size_t)(row - N_HEADS * BATCH) * HEAD_DIM + n] = c[i];
    }
}

// ---------------------------------------------------------------------------
// Attention core: one 128-thread block per (b, kv).
//   4 GQA queries x padded_len keys, new k/v substituted at current_pos.
// ---------------------------------------------------------------------------
__global__ __launch_bounds__(128)
void attn_kernel(const float* __restrict__ q_rope, const float* __restrict__ k_new,
                 const float* __restrict__ xqkv,
                 const float* __restrict__ cache_k, const float* __restrict__ cache_v,
                 const float* __restrict__ mask,
                 const int* __restrict__ sp_ptr, const int* __restrict__ cp_ptr,
                 float* __restrict__ attn_out) {
    __shared__ float sQ[G * HEAD_DIM];       // 2 KB
    __shared__ float sS[G * CACHE_LEN];      // 8 KB (scores -> probs)

    const int tid = threadIdx.x;             // 0..127
    const int b   = blockIdx.x >> 3;
    const int kv  = blockIdx.x & 7;

    const int sp   = sp_ptr[0];
    const int cp   = cp_ptr[0];
    int plen = ((sp + 1 + 31) >> 5) << 5;    // nearest_32(start_pos+1)
    if (plen > 4096) plen = 4096;            // sliding window
    if (plen > CACHE_LEN) plen = CACHE_LEN;  // cache extent

    // stage the 4 rotated queries for this (kv, b)
    #pragma unroll
    for (int g = 0; g < G; ++g)
        sQ[g * HEAD_DIM + tid] =
            q_rope[(size_t)((kv * G + g) * BATCH + b) * HEAD_DIM + tid];
    __syncthreads();

    // ---- scores: each thread owns positions t = tid, tid+128, ... ----------
    const float scale = 0.08838834764831845f;     // 1/sqrt(128)
    const size_t cbase = ((size_t)b * N_KV + kv) * CACHE_LEN;
    for (int t = tid; t < plen; t += 128) {
        const float* kvec = (t == cp)
            ? (k_new + (size_t)(kv * BATCH + b) * HEAD_DIM)
            : (cache_k + (cbase + t) * HEAD_DIM);
        float d0 = 0.f, d1 = 0.f, d2 = 0.f, d3 = 0.f;
        #pragma unroll 4
        for (int d = 0; d < HEAD_DIM; ++d) {
            float kd = kvec[d];
            d0 += sQ[d]                * kd;
            d1 += sQ[HEAD_DIM + d]     * kd;
            d2 += sQ[2 * HEAD_DIM + d] * kd;
            d3 += sQ[3 * HEAD_DIM + d] * kd;
        }
        float sc[G] = {d0, d1, d2, d3};
        #pragma unroll
        for (int g = 0; g < G; ++g)
            sS[g * CACHE_LEN + t] = sc[g] * scale +
                mask[(size_t)((kv * G + g) * BATCH + b) * plen + t];
    }
    __syncthreads();

    // ---- softmax: wave g (of 4) normalizes row g ---------------------------
    {
        const int g    = tid >> 5;
        const int lane = tid & 31;
        float* row = sS + g * CACHE_LEN;
        float m = -3.402823466e+38f;
        for (int t = lane; t < plen; t += 32) m = fmaxf(m, row[t]);
        #pragma unroll
        for (int off = 16; off >= 1; off >>= 1) m = fmaxf(m, __shfl_xor(m, off, 32));
        float s = 0.f;
        for (int t = lane; t < plen; t += 32) {
            float e = __expf(row[t] - m);
            row[t] = e;
            s += e;
        }
        #pragma unroll
        for (int off = 16; off >= 1; off >>= 1) s += __shfl_xor(s, off, 32);
        float inv = 1.f / s;
        for (int t = lane; t < plen; t += 32) row[t] *= inv;
    }
    __syncthreads();

    // ---- out = probs @ V : thread owns one head dim d ----------------------
    {
        const int d = tid;
        const float vcp = xqkv[(size_t)b * QKV_F + V_OFF + kv * HEAD_DIM + d];
        float a0 = 0.f, a1 = 0.f, a2 = 0.f, a3 = 0.f;
        for (int t = 0; t < plen; ++t) {
            float v = (t == cp) ? vcp : cache_v[(cbase + t) * HEAD_DIM + d];
            a0 += sS[t]                 * v;
            a1 += sS[CACHE_LEN + t]     * v;
            a2 += sS[2 * CACHE_LEN + t] * v;
            a3 += sS[3 * CACHE_LEN + t] * v;
        }
        float acc[G] = {a0, a1, a2, a3};
        #pragma unroll
        for (int g = 0; g < G; ++g)
            attn_out[(size_t)b * HIDDEN + (kv * G + g) * HEAD_DIM + d] = acc[g];
    }
}

// ---------------------------------------------------------------------------
// Host-side launcher (all on `stream`, graph-capture safe)
// ---------------------------------------------------------------------------
extern "C" void kernel_launch(void* const* d_in, const int* in_sizes, int n_in,
                              void* d_out, int out_size, void* d_ws, size_t ws_size,
                              hipStream_t stream) {
    const float* x       = (const float*)d_in[0];   // (32, 4096)
    const float* wqkv    = (const float*)d_in[1];   // (4096, 6144)
    const float* wo      = (const float*)d_in[2];   // (4096, 4096)
    const float* rot     = (const float*)d_in[3];   // (128, 128)
    const float* cache_k = (const float*)d_in[4];   // (32, 8, 512, 128)
    const float* cache_v = (const float*)d_in[5];
    const float* mask    = (const float*)d_in[6];   // (1, 32, 32, plen)
    const int*   sp      = (const int*)d_in[7];
    const int*   cp      = (const int*)d_in[8];
    float*       out     = (float*)d_out;           // (32, 4096)

    float* ws    = (float*)d_ws;
    float* xqkv  = ws;                                        // 32*6144
    float* qrope = xqkv  + (size_t)BATCH * QKV_F;             // 1024*128
    float* knew  = qrope + (size_t)N_HEADS * BATCH * HEAD_DIM;// 256*128
    float* attn  = knew  + (size_t)N_KV * BATCH * HEAD_DIM;   // 32*4096

    dim3 blk(32, WAVES_N, 1);

    // 1) xqkv = x @ wqkv                 M=32, N=6144, K=4096  (wqkv read once)
    wmma_gemm_f32_m32<<<dim3(QKV_F / (16 * WAVES_N)), blk, 0, stream>>>(
        x, wqkv, xqkv, QKV_F, HIDDEN, HIDDEN, QKV_F, QKV_F);

    // 2) RoPE for q (1024 rows) and k (256 rows): 80 m-tiles
    rope_wmma<<<dim3((N_HEADS * BATCH + N_KV * BATCH) / 16), blk, 0, stream>>>(
        xqkv, rot, qrope, knew);

    // 3) attention core: one block per (b, kv)
    attn_kernel<<<dim3(BATCH * N_KV), dim3(128), 0, stream>>>(
        qrope, knew, xqkv, cache_k, cache_v, mask, sp, cp, attn);

    // 4) dense_out = attn @ wo           M=32, N=4096, K=4096  (wo read once)
    wmma_gemm_f32_m32<<<dim3(HIDDEN / (16 * WAVES_N)), blk, 0, stream>>>(
        attn, wo, out, HIDDEN, HIDDEN, HIDDEN, HIDDEN, HIDDEN);
}